// Heteroconv_47794396070092
// MI455X (gfx1250) — compile-verified
//
#include <hip/hip_runtime.h>
#include <hip/hip_bf16.h>
#include <math.h>

// ---------------------------------------------------------------------------
// HeteroConv (6x GAT + 1x SAGE) for MI455X / gfx1250, wave32.
// GEMMs (hs = x @ Ws, SAGE linears) use V_WMMA_F32_16X16X4_F32 (exact f32).
// Attention scalars use the identity e = x_src @ (Ws@as) (no hd needed).
// Edge softmax: atomicMax on order-preserving u32 keys, then atomic f32 adds.
// ---------------------------------------------------------------------------

typedef __attribute__((ext_vector_type(2))) float v2f;
typedef __attribute__((ext_vector_type(8))) float v8f;

#define HD 128  // hidden dim

static __device__ __forceinline__ float lrelu(float x) { return x > 0.f ? x : 0.2f * x; }

// order-preserving float<->uint (for atomicMax over signed floats)
static __device__ __forceinline__ unsigned f2ord(float f) {
  unsigned u = __float_as_uint(f);
  return (u & 0x80000000u) ? ~u : (u | 0x80000000u);
}
static __device__ __forceinline__ float ord2f(unsigned k) {
  return __uint_as_float((k & 0x80000000u) ? (k & 0x7FFFFFFFu) : ~k);
}

// ---------------------------------------------------------------------------
// WMMA GEMM: Cout[M x 128] = A[M x K] @ B[K x 128] (+ C0) (+ bias)
// Block = 256 threads = 8 waves; wave w owns the 16x16 tile at cols w*16.
// Grid.x = M/16 (M is always a multiple of 16 here -> EXEC all-ones).
// A-frag (16x4 f32): lanes 0-15 carry K=k..k+1, lanes 16-31 carry K=k+2..k+3.
// B-frag (4x16 f32): vgpr0 = row k (k+2 for hi lanes), vgpr1 = row k+1 (k+3).
// ---------------------------------------------------------------------------
__global__ __launch_bounds__(256) void wmma_gemm_kernel(
    const float* __restrict__ A, const float* __restrict__ B,
    const float* __restrict__ C0, const float* __restrict__ bias,
    float* __restrict__ Cout, int K) {
  const int lane = threadIdx.x & 31;
  const int wave = threadIdx.x >> 5;
  const int m0 = blockIdx.x << 4;
  const int n0 = wave << 4;
  const int hi = lane >> 4;   // 0: lanes 0-15, 1: lanes 16-31
  const int l15 = lane & 15;

  const float* Arow = A + (size_t)(m0 + l15) * (size_t)K;
  v8f acc = {0.f, 0.f, 0.f, 0.f, 0.f, 0.f, 0.f, 0.f};

  for (int k = 0; k < K; k += 4) {
    const int ka = k + (hi << 1);
    v2f afrag;
    afrag.x = Arow[ka];
    afrag.y = Arow[ka + 1];
    v2f bfrag;
    bfrag.x = B[(size_t)ka * HD + n0 + l15];
    bfrag.y = B[(size_t)(ka + 1) * HD + n0 + l15];
    acc = __builtin_amdgcn_wmma_f32_16x16x4_f32(false, afrag, false, bfrag,
                                                (short)0, acc, false, false);
  }

#pragma unroll
  for (int r = 0; r < 8; ++r) {
    const int row = m0 + r + (hi << 3);
    const int col = n0 + l15;
    const size_t idx = (size_t)row * HD + col;
    float v = acc[r];
    if (C0) v += C0[idx];
    if (bias) v += bias[col];
    Cout[idx] = v;
  }
}

// out[i] = dot(W[i, 0:128], v)   (tiny: W is F x 128, F <= 256)
__global__ void matvec_kernel(const float* __restrict__ W, const float* __restrict__ v,
                              float* __restrict__ out, int F) {
  int i = blockIdx.x * blockDim.x + threadIdx.x;
  if (i >= F) return;
  const float* row = W + (size_t)i * HD;
  float s = 0.f;
#pragma unroll 4
  for (int j = 0; j < HD; ++j) s += row[j] * v[j];
  out[i] = s;
}

// out[n] = dot(X[n, 0:F], w)  -- one wave per node, float4 lanes + shuffle reduce
__global__ void node_dot_kernel(const float* __restrict__ X, const float* __restrict__ w,
                                float* __restrict__ out, int N, int F) {
  int gid = blockIdx.x * blockDim.x + threadIdx.x;
  int node = gid >> 5;
  int lane = threadIdx.x & 31;
  if (node >= N) return;
  const float* xr = X + (size_t)node * F;
  float s = 0.f;
  for (int t = lane * 4; t < F; t += 128) {
    float4 x = *(const float4*)(xr + t);
    float4 ww = *(const float4*)(w + t);
    s += x.x * ww.x + x.y * ww.y + x.z * ww.z + x.w * ww.w;
  }
  for (int off = 16; off > 0; off >>= 1) s += __shfl_down(s, off);
  if (lane == 0) out[node] = s;
}

__global__ void bias2_init_kernel(float* __restrict__ out, const float* __restrict__ b1,
                                  const float* __restrict__ b2, int N) {
  int i = blockIdx.x * blockDim.x + threadIdx.x;
  if (i >= N * HD) return;
  int col = i & (HD - 1);
  out[i] = b1[col] + b2[col];
}

__global__ void softmax_init_kernel(unsigned* __restrict__ menc, float* __restrict__ den, int N) {
  int i = blockIdx.x * blockDim.x + threadIdx.x;
  if (i >= N) return;
  menc[i] = 0x007FFFFFu;  // f2ord(-inf)
  den[i] = 0.f;
}

__global__ void zero_kernel(float* __restrict__ p, int n) {
  int i = blockIdx.x * blockDim.x + threadIdx.x;
  if (i < n) p[i] = 0.f;
}

// pass 1: segment max of leaky_relu(e[src] + f[dst])
__global__ void edge_max_kernel(const int* __restrict__ src, const int* __restrict__ dst,
                                const float* __restrict__ e, const float* __restrict__ f,
                                unsigned* __restrict__ menc, int E) {
  int i = blockIdx.x * blockDim.x + threadIdx.x;
  if (i >= E) return;
  float a = lrelu(e[src[i]] + f[dst[i]]);
  atomicMax(menc + dst[i], f2ord(a));
}

// pass 2: segment sum of exp(a - m[dst])
__global__ void edge_den_kernel(const int* __restrict__ src, const int* __restrict__ dst,
                                const float* __restrict__ e, const float* __restrict__ f,
                                const unsigned* __restrict__ menc, float* __restrict__ den, int E) {
  int i = blockIdx.x * blockDim.x + threadIdx.x;
  if (i >= E) return;
  int d = dst[i];
  float a = lrelu(e[src[i]] + f[d]);
  atomicAdd(den + d, __expf(a - ord2f(menc[d])));
}

// pass 3: out[dst] += hs[src] * alpha   (one wave per edge, float4 per lane)
__global__ void edge_scatter_kernel(const int* __restrict__ src, const int* __restrict__ dst,
                                    const float* __restrict__ e, const float* __restrict__ f,
                                    const unsigned* __restrict__ menc, const float* __restrict__ den,
                                    const float* __restrict__ hs, float* __restrict__ out, int E) {
  int gid = blockIdx.x * blockDim.x + threadIdx.x;
  int eidx = gid >> 5;
  int lane = threadIdx.x & 31;
  if (eidx >= E) return;
  int s = src[eidx], d = dst[eidx];
  float a = lrelu(e[s] + f[d]);
  float alpha = __expf(a - ord2f(menc[d])) / (den[d] + 1e-16f);
  float4 h = *(const float4*)(hs + (size_t)s * HD + lane * 4);
  float* o = out + (size_t)d * HD + lane * 4;
  atomicAdd(o + 0, h.x * alpha);
  atomicAdd(o + 1, h.y * alpha);
  atomicAdd(o + 2, h.z * alpha);
  atomicAdd(o + 3, h.w * alpha);
}

// SAGE: agg[dst] += x[src]; cnt[dst] += 1   (one wave per edge)
__global__ void sage_scatter_kernel(const int* __restrict__ src, const int* __restrict__ dst,
                                    const float* __restrict__ X, float* __restrict__ agg,
                                    float* __restrict__ cnt, int E) {
  int gid = blockIdx.x * blockDim.x + threadIdx.x;
  int eidx = gid >> 5;
  int lane = threadIdx.x & 31;
  if (eidx >= E) return;
  int s = src[eidx], d = dst[eidx];
  float4 x = *(const float4*)(X + (size_t)s * HD + lane * 4);
  float* o = agg + (size_t)d * HD + lane * 4;
  atomicAdd(o + 0, x.x);
  atomicAdd(o + 1, x.y);
  atomicAdd(o + 2, x.z);
  atomicAdd(o + 3, x.w);
  if (lane == 0) atomicAdd(cnt + d, 1.0f);
}

__global__ void sage_div_kernel(float* __restrict__ agg, const float* __restrict__ cnt, int total) {
  int i = blockIdx.x * blockDim.x + threadIdx.x;
  if (i >= total) return;
  agg[i] = agg[i] / fmaxf(cnt[i >> 7], 1.0f);
}

// ---------------------------------------------------------------------------
static inline int cdiv(int a, int b) { return (a + b - 1) / b; }

extern "C" void kernel_launch(void* const* d_in, const int* in_sizes, int n_in,
                              void* d_out, int out_size, void* d_ws, size_t ws_size,
                              hipStream_t stream) {
  (void)n_in; (void)out_size; (void)ws_size;
  // ---- inputs (setup_inputs dict insertion order, params nested in REL order)
  const float* x_drug = (const float*)d_in[0];   // 20000 x 128
  const float* x_prot = (const float*)d_in[1];   // 20000 x 256
  const float* x_path = (const float*)d_in[2];   // 5000 x 128  (only f needed)
  const float* x_dis  = (const float*)d_in[3];   // 5000 x 128  (only f needed)
  const int* dp_src   = (const int*)d_in[4];   const int* dp_dst   = (const int*)d_in[5];
  const int* ddi_src  = (const int*)d_in[6];   const int* ddi_dst  = (const int*)d_in[7];
  const int* ppa_src  = (const int*)d_in[8];   const int* ppa_dst  = (const int*)d_in[9];
  const int* dpa_src  = (const int*)d_in[10];  const int* dpa_dst  = (const int*)d_in[11];
  const int* pd_src   = (const int*)d_in[12];  const int* pd_dst   = (const int*)d_in[13];
  const int* ppi_src  = (const int*)d_in[14];  const int* ppi_dst  = (const int*)d_in[15];
  const int* ddis_src = (const int*)d_in[16];  const int* ddis_dst = (const int*)d_in[17];
  const int E_dp = in_sizes[4], E_ddi = in_sizes[6], E_ppa = in_sizes[8], E_dpa = in_sizes[10];
  const int E_pd = in_sizes[12], E_ppi = in_sizes[14], E_ddis = in_sizes[16];
  // params: dp(GAT) ddi(SAGE) ppa dpa pd ppi ddis; GAT={Ws,Wd,as,ad,b}, SAGE={Wl,bl,Wr}
  const float* dp_Ws = (const float*)d_in[18];  const float* dp_Wd = (const float*)d_in[19];
  const float* dp_as = (const float*)d_in[20];  const float* dp_ad = (const float*)d_in[21];
  const float* dp_b  = (const float*)d_in[22];
  const float* ddi_Wl = (const float*)d_in[23]; const float* ddi_bl = (const float*)d_in[24];
  const float* ddi_Wr = (const float*)d_in[25];
  const float* ppa_Ws = (const float*)d_in[26]; const float* ppa_Wd = (const float*)d_in[27];
  const float* ppa_as = (const float*)d_in[28]; const float* ppa_ad = (const float*)d_in[29];
  const float* ppa_b  = (const float*)d_in[30];
  const float* dpa_Ws = (const float*)d_in[31]; const float* dpa_Wd = (const float*)d_in[32];
  const float* dpa_as = (const float*)d_in[33]; const float* dpa_ad = (const float*)d_in[34];
  const float* dpa_b  = (const float*)d_in[35];
  const float* pd_Ws = (const float*)d_in[36];  const float* pd_Wd = (const float*)d_in[37];
  const float* pd_as = (const float*)d_in[38];  const float* pd_ad = (const float*)d_in[39];
  const float* pd_b  = (const float*)d_in[40];
  const float* ppi_Ws = (const float*)d_in[41]; const float* ppi_Wd = (const float*)d_in[42];
  const float* ppi_as = (const float*)d_in[43]; const float* ppi_ad = (const float*)d_in[44];
  const float* ppi_b  = (const float*)d_in[45];
  const float* ddis_Ws = (const float*)d_in[46]; const float* ddis_Wd = (const float*)d_in[47];
  const float* ddis_as = (const float*)d_in[48]; const float* ddis_ad = (const float*)d_in[49];
  const float* ddis_b  = (const float*)d_in[50];

  // ---- outputs: concat(drug 20000x128, prot 20000x128, path 5000x128, dis 5000x128)
  float* out = (float*)d_out;
  float* out_drug = out;
  float* out_prot = out + 2560000;
  float* out_path = out + 5120000;
  float* out_dis  = out + 5760000;

  // ---- workspace layout (floats): ~21 MB
  float* ws = (float*)d_ws;
  float* hs   = ws;                 // 20000*128 = 2,560,000
  float* agg  = ws + 2560000;       // 20000*128 = 2,560,000
  float* ebuf = ws + 5120000;       // 20,000
  float* fbuf = ebuf + 20000;       // 20,000
  unsigned* menc = (unsigned*)(fbuf + 20000);   // 20,000
  float* den  = (float*)(menc + 20000);         // 20,000 (also SAGE cnt)
  float* vecA = den + 20000;        // 256 (Ws @ as)
  float* vecB = vecA + 256;         // 256 (Wd @ ad)

  // ---- init output with summed HeteroConv biases (edgeless dst nodes -> bias only)
  bias2_init_kernel<<<cdiv(20000 * HD, 256), 256, 0, stream>>>(out_prot, dp_b, ppi_b, 20000);
  bias2_init_kernel<<<cdiv(5000 * HD, 256), 256, 0, stream>>>(out_path, ppa_b, dpa_b, 5000);
  bias2_init_kernel<<<cdiv(5000 * HD, 256), 256, 0, stream>>>(out_dis, pd_b, ddis_b, 5000);

  // ---- SAGE (ddi: drug -> drug): mean-agg, then out = agg@Wl + bl + x@Wr
  float* cnt = den;
  zero_kernel<<<cdiv(2560000, 256), 256, 0, stream>>>(agg, 2560000);
  zero_kernel<<<cdiv(20000, 256), 256, 0, stream>>>(cnt, 20000);
  sage_scatter_kernel<<<cdiv(E_ddi * 32, 256), 256, 0, stream>>>(ddi_src, ddi_dst, x_drug, agg, cnt, E_ddi);
  sage_div_kernel<<<cdiv(2560000, 256), 256, 0, stream>>>(agg, cnt, 2560000);
  wmma_gemm_kernel<<<20000 / 16, 256, 0, stream>>>(agg, ddi_Wl, nullptr, nullptr, hs, 128);
  wmma_gemm_kernel<<<20000 / 16, 256, 0, stream>>>(x_drug, ddi_Wr, hs, ddi_bl, out_drug, 128);

  // ---- GAT relation runner (sequential on stream; buffers reused safely)
  auto run_gat = [&](const float* xs, int Ns, int Fs, const float* xd, int Nd, int Fd,
                     const int* src, const int* dst, int E,
                     const float* Wsm, const float* Wdm, const float* asv, const float* adv,
                     float* out_seg) {
    matvec_kernel<<<cdiv(Fs, 256), 256, 0, stream>>>(Wsm, asv, vecA, Fs);
    matvec_kernel<<<cdiv(Fd, 256), 256, 0, stream>>>(Wdm, adv, vecB, Fd);
    node_dot_kernel<<<cdiv(Ns * 32, 256), 256, 0, stream>>>(xs, vecA, ebuf, Ns, Fs);
    node_dot_kernel<<<cdiv(Nd * 32, 256), 256, 0, stream>>>(xd, vecB, fbuf, Nd, Fd);
    wmma_gemm_kernel<<<Ns / 16, 256, 0, stream>>>(xs, Wsm, nullptr, nullptr, hs, Fs);
    softmax_init_kernel<<<cdiv(Nd, 256), 256, 0, stream>>>(menc, den, Nd);
    edge_max_kernel<<<cdiv(E, 256), 256, 0, stream>>>(src, dst, ebuf, fbuf, menc, E);
    edge_den_kernel<<<cdiv(E, 256), 256, 0, stream>>>(src, dst, ebuf, fbuf, menc, den, E);
    edge_scatter_kernel<<<cdiv(E * 32, 256), 256, 0, stream>>>(src, dst, ebuf, fbuf, menc, den,
                                                               hs, out_seg, E);
  };

  // prot_out += GAT(dp: drug->prot) + GAT(ppi: prot->prot)
  run_gat(x_drug, 20000, 128, x_prot, 20000, 256, dp_src, dp_dst, E_dp,
          dp_Ws, dp_Wd, dp_as, dp_ad, out_prot);
  run_gat(x_prot, 20000, 256, x_prot, 20000, 256, ppi_src, ppi_dst, E_ppi,
          ppi_Ws, ppi_Wd, ppi_as, ppi_ad, out_prot);
  // path_out += GAT(ppa: prot->path) + GAT(dpa: drug->path)
  run_gat(x_prot, 20000, 256, x_path, 5000, 128, ppa_src, ppa_dst, E_ppa,
          ppa_Ws, ppa_Wd, ppa_as, ppa_ad, out_path);
  run_gat(x_drug, 20000, 128, x_path, 5000, 128, dpa_src, dpa_dst, E_dpa,
          dpa_Ws, dpa_Wd, dpa_as, dpa_ad, out_path);
  // dis_out += GAT(pd: prot->dis) + GAT(ddis: drug->dis)
  run_gat(x_prot, 20000, 256, x_dis, 5000, 128, pd_src, pd_dst, E_pd,
          pd_Ws, pd_Wd, pd_as, pd_ad, out_dis);
  run_gat(x_drug, 20000, 128, x_dis, 5000, 128, ddis_src, ddis_dst, E_ddis,
          ddis_Ws, ddis_Wd, ddis_as, ddis_ad, out_dis);
}